// AttentionPairBias_88115549045150
// MI455X (gfx1250) — compile-verified
//
#include <hip/hip_runtime.h>
#include <hip/hip_bf16.h>
#include <math.h>
#include <stdint.h>

// Problem constants (match reference)
constexpr int Bc  = 2;
constexpr int Nc  = 512;
constexpr int DSc = 512;
constexpr int DZc = 128;
constexpr int Hc  = 16;
constexpr int HDc = 32;

typedef __attribute__((ext_vector_type(2))) float v2f;
typedef __attribute__((ext_vector_type(8))) float v8f;

__device__ __forceinline__ float wave_sum(float x) {
    #pragma unroll
    for (int m = 16; m > 0; m >>= 1) x += __shfl_xor(x, m, 32);
    return x;
}
__device__ __forceinline__ float wave_max(float x) {
    #pragma unroll
    for (int m = 16; m > 0; m >>= 1) x = fmaxf(x, __shfl_xor(x, m, 32));
    return x;
}

// --- CDNA5 async global->LDS DMA (ASYNCcnt-tracked), 16B per lane ---------
// LDS operand is the AS3 byte offset = low 32 bits of a generic pointer to
// __shared__ (flat->LDS aperture rule).
__device__ __forceinline__ void async_g2l_b128(void* lds, const void* gaddr) {
    const uint32_t loff = (uint32_t)(uintptr_t)lds;
    asm volatile("global_load_async_to_lds_b128 %0, %1, off"
                 :: "v"(loff), "v"(gaddr)
                 : "memory");
}
__device__ __forceinline__ void wait_async0() {
#if __has_builtin(__builtin_amdgcn_s_wait_asynccnt)
    __builtin_amdgcn_s_wait_asynccnt(0);
#else
    asm volatile("s_wait_asynccnt 0x0" ::: "memory");
#endif
}

// ---------------------------------------------------------------------------
// Generic 1024x512 @ 512x512 GEMM, one wave computes a 16x16 tile via
// v_wmma_f32_16x16x4_f32.  MODE 1: store permuted [B,H,N,HD] (QKV);
// MODE 2: sigmoid, row-major (gate); MODE 3: gate * (x+b), row-major (final).
// ---------------------------------------------------------------------------
template <int MODE>
__global__ __launch_bounds__(32) void gemm512_k(
    const float* __restrict__ A, const float* __restrict__ W,
    const float* __restrict__ bias, const float* __restrict__ gate,
    float* __restrict__ out)
{
    const int lane = threadIdx.x;
    const int half = lane >> 4;      // K-pair select per ISA A/B layout
    const int l16  = lane & 15;
    const int m0 = blockIdx.x * 16;  // row tile   (M = B*N = 1024)
    const int c0 = blockIdx.y * 16;  // col tile   (512 cols)

    const float* arow = A + (size_t)(m0 + l16) * DSc;
    v8f acc = {};
    for (int kk = 0; kk < DSc; kk += 4) {
        const int ka = kk + 2 * half;
        v2f a, b;
        a.x = arow[ka];
        a.y = arow[ka + 1];
        b.x = W[(size_t)ka * DSc + c0 + l16];
        b.y = W[(size_t)(ka + 1) * DSc + c0 + l16];
        acc = __builtin_amdgcn_wmma_f32_16x16x4_f32(
            false, a, false, b, (short)0, acc, false, false);
    }

    const float bv = bias[c0 + l16];
    #pragma unroll
    for (int j = 0; j < 8; ++j) {
        const int gm = m0 + j + 8 * half;  // global row  (D layout: M = j / j+8)
        const int gc = c0 + l16;           // global col  (D layout: N = lane%16)
        float v = acc[j] + bv;
        if constexpr (MODE == 1) {
            const int b_ = gm >> 9, n_ = gm & (Nc - 1);
            const int h_ = gc >> 5, hd = gc & (HDc - 1);
            out[((size_t)(b_ * Hc + h_) * Nc + n_) * HDc + hd] = v;
        } else if constexpr (MODE == 2) {
            out[(size_t)gm * DSc + gc] = 1.0f / (1.0f + __expf(-v));
        } else {
            out[(size_t)gm * DSc + gc] = gate[(size_t)gm * DSc + gc] * v;
        }
    }
}

// ---------------------------------------------------------------------------
// Pair bias: LayerNorm(z) @ Wp -> bias[B,H,N,N].
// One wave per (b, q, 16-wide k tile): 16 rows x K=128 x 16 heads.
// z tile DMA'd into LDS via global_load_async_to_lds_b128 (stride-132 pad =>
// conflict-free fragment reads); output transposed through LDS so the
// [B,H,N,N] writes go out as contiguous 64B segments per head.
// ---------------------------------------------------------------------------
__global__ __launch_bounds__(32) void pair_bias_k(
    const float* __restrict__ z, const float* __restrict__ ln_g,
    const float* __restrict__ ln_b, const float* __restrict__ Wp,
    float* __restrict__ bias)
{
    __shared__ float zsh[16 * 132];
    __shared__ float wpS[DZc * Hc];
    __shared__ float gS[DZc], bS[DZc];
    __shared__ float muS[16], rsS[16];
    __shared__ float tile[256];

    const int lane = threadIdx.x;
    const int half = lane >> 4;
    const int l16  = lane & 15;

    const int id = blockIdx.x;          // B * N * (N/16) = 32768 blocks
    const int kt = id & 31;
    const int q  = (id >> 5) & (Nc - 1);
    const int b_ = id >> 14;

    const float* zrow = z + ((((size_t)b_ * Nc + q) * Nc) + kt * 16) * DZc;

    // Async DMA: 16 rows x 128 floats, 32 B128 chunks per row, 16 per lane.
    #pragma unroll
    for (int i = 0; i < 16; ++i) {
        const int c  = i * 32 + lane;   // chunk id
        const int r  = c >> 5;          // row (32 chunks per row)
        const int c4 = (c & 31) * 4;    // float column
        async_g2l_b128(&zsh[r * 132 + c4], zrow + r * 128 + c4);
    }
    for (int i = lane; i < DZc * Hc; i += 32) wpS[i] = Wp[i];
    for (int i = lane; i < DZc; i += 32) { gS[i] = ln_g[i]; bS[i] = ln_b[i]; }
    wait_async0();
    __syncthreads();

    // LayerNorm stats, one row per wave-pass
    for (int r = 0; r < 16; ++r) {
        float s = 0.f, s2 = 0.f;
        for (int i = lane; i < DZc; i += 32) {
            const float x = zsh[r * 132 + i];
            s += x; s2 += x * x;
        }
        s = wave_sum(s); s2 = wave_sum(s2);
        const float mu  = s * (1.0f / DZc);
        const float var = s2 * (1.0f / DZc) - mu * mu;
        if (lane == 0) { muS[r] = mu; rsS[r] = rsqrtf(var + 1e-5f); }
    }
    __syncthreads();

    const float mu = muS[l16];
    const float rs = rsS[l16];
    v8f acc = {};
    for (int kk = 0; kk < DZc; kk += 4) {
        const int ka = kk + 2 * half;
        v2f a, b;
        a.x = (zsh[l16 * 132 + ka]     - mu) * rs * gS[ka]     + bS[ka];
        a.y = (zsh[l16 * 132 + ka + 1] - mu) * rs * gS[ka + 1] + bS[ka + 1];
        b.x = wpS[ka * Hc + l16];
        b.y = wpS[(ka + 1) * Hc + l16];
        acc = __builtin_amdgcn_wmma_f32_16x16x4_f32(
            false, a, false, b, (short)0, acc, false, false);
    }

    // transpose 16(k) x 16(h) through LDS -> contiguous writes along k per head
    #pragma unroll
    for (int j = 0; j < 8; ++j) tile[(j + 8 * half) * 16 + l16] = acc[j];
    __syncthreads();
    for (int i = lane; i < 256; i += 32) {
        const int h_ = i >> 4, m = i & 15;
        bias[(((size_t)(b_ * Hc + h_) * Nc) + q) * Nc + kt * 16 + m] =
            tile[m * 16 + h_];
    }
}

// ---------------------------------------------------------------------------
// Attention: one wave per (b, h, 16-query tile).
// S = QK^T*scale + bias (masked) held in LDS [16][516]; wave-parallel
// softmax; PV with dual accumulators (HD=32 = two 16-col tiles).
// ---------------------------------------------------------------------------
__global__ __launch_bounds__(32) void attn_k(
    const float* __restrict__ Q, const float* __restrict__ K,
    const float* __restrict__ V, const float* __restrict__ bias,
    const unsigned char* __restrict__ mask, float* __restrict__ outw)
{
    __shared__ float S[16 * 516];
    __shared__ float sumS[16];

    const int lane = threadIdx.x;
    const int half = lane >> 4;
    const int l16  = lane & 15;

    const int id = blockIdx.x;          // B*H*(N/16) = 1024 blocks
    const int qt = id & 31;
    const int h_ = (id >> 5) & (Hc - 1);
    const int b_ = id >> 9;
    const int q0 = qt * 16;

    const size_t bh = (size_t)(b_ * Hc + h_);
    const float* Qp = Q + bh * Nc * HDc;
    const float* Kp = K + bh * Nc * HDc;
    const float* Vp = V + bh * Nc * HDc;
    const float* Bp = bias + bh * Nc * Nc;
    const float scale = 0.17677669529663687f;   // 1/sqrt(HD)

    const float* qrow = Qp + (size_t)(q0 + l16) * HDc;

    // --- scores ---
    for (int kt = 0; kt < Nc / 16; ++kt) {
        v8f acc = {};
        #pragma unroll
        for (int kk = 0; kk < HDc; kk += 4) {
            const int ka = kk + 2 * half;
            v2f a, b;
            a.x = qrow[ka];
            a.y = qrow[ka + 1];
            const float* krow = Kp + (size_t)(kt * 16 + l16) * HDc + ka;
            b.x = krow[0];
            b.y = krow[1];
            acc = __builtin_amdgcn_wmma_f32_16x16x4_f32(
                false, a, false, b, (short)0, acc, false, false);
        }
        #pragma unroll
        for (int j = 0; j < 8; ++j) {
            const int m = j + 8 * half;          // query row in tile
            const int n = kt * 16 + l16;         // key index
            float v = acc[j] * scale + Bp[(size_t)(q0 + m) * Nc + n];
            if (!mask[b_ * Nc + n]) v = -1.0e30f;
            S[m * 516 + n] = v;
        }
    }
    __syncthreads();

    // --- softmax per row (normalization folded into PV epilogue) ---
    for (int r = 0; r < 16; ++r) {
        float mx = -3.0e38f;
        #pragma unroll
        for (int i = 0; i < 16; ++i) mx = fmaxf(mx, S[r * 516 + i * 32 + lane]);
        mx = wave_max(mx);
        float sm = 0.f;
        #pragma unroll
        for (int i = 0; i < 16; ++i) {
            const int c = r * 516 + i * 32 + lane;
            const float e = __expf(S[c] - mx);
            S[c] = e;
            sm += e;
        }
        sm = wave_sum(sm);
        if (lane == 0) sumS[r] = sm;
    }
    __syncthreads();

    // --- P @ V ---
    v8f acc0 = {}, acc1 = {};
    for (int kk = 0; kk < Nc; kk += 4) {
        const int ka = kk + 2 * half;
        v2f a, b0, b1;
        a.x = S[l16 * 516 + ka];
        a.y = S[l16 * 516 + ka + 1];
        const float* vr0 = Vp + (size_t)ka * HDc;
        const float* vr1 = Vp + (size_t)(ka + 1) * HDc;
        b0.x = vr0[l16];      b0.y = vr1[l16];
        b1.x = vr0[16 + l16]; b1.y = vr1[16 + l16];
        acc0 = __builtin_amdgcn_wmma_f32_16x16x4_f32(
            false, a, false, b0, (short)0, acc0, false, false);
        acc1 = __builtin_amdgcn_wmma_f32_16x16x4_f32(
            false, a, false, b1, (short)0, acc1, false, false);
    }

    #pragma unroll
    for (int j = 0; j < 8; ++j) {
        const int m  = j + 8 * half;
        const float inv = 1.0f / sumS[m];
        const int qq = q0 + m;
        const size_t base = (((size_t)b_ * Nc + qq) * Hc + h_) * (size_t)HDc;
        outw[base + l16]      = acc0[j] * inv;   // attn_out layout [B,N,H,HD]
        outw[base + 16 + l16] = acc1[j] * inv;
    }
}

// ---------------------------------------------------------------------------
// Launch
// ---------------------------------------------------------------------------
extern "C" void kernel_launch(void* const* d_in, const int* in_sizes, int n_in,
                              void* d_out, int out_size, void* d_ws, size_t ws_size,
                              hipStream_t stream) {
    const float* s    = (const float*)d_in[0];
    const float* z    = (const float*)d_in[1];
    const unsigned char* mask = (const unsigned char*)d_in[2];
    const float* Wq = (const float*)d_in[3];  const float* bq = (const float*)d_in[4];
    const float* Wk = (const float*)d_in[5];  const float* bk = (const float*)d_in[6];
    const float* Wv = (const float*)d_in[7];  const float* bv = (const float*)d_in[8];
    const float* lg = (const float*)d_in[9];  const float* lb = (const float*)d_in[10];
    const float* Wp = (const float*)d_in[11];
    const float* Wg = (const float*)d_in[12]; const float* bg = (const float*)d_in[13];
    const float* Wo = (const float*)d_in[14]; const float* bo = (const float*)d_in[15];
    float* out = (float*)d_out;

    // workspace carve-up (floats)
    float* ws = (float*)d_ws;
    const size_t SZ = (size_t)Bc * Nc * DSc;        // 524288
    float* qb = ws;
    float* kb = qb + SZ;
    float* vb = kb + SZ;
    float* gb = vb + SZ;
    float* pb = gb + SZ;                            // bias: B*H*N*N = 8388608
    float* ob = pb + (size_t)Bc * Hc * Nc * Nc;

    const dim3 gg((Bc * Nc) / 16, DSc / 16);        // 64 x 32 tiles, 1 wave each
    gemm512_k<1><<<gg, 32, 0, stream>>>(s, Wq, bq, nullptr, qb);
    gemm512_k<1><<<gg, 32, 0, stream>>>(s, Wk, bk, nullptr, kb);
    gemm512_k<1><<<gg, 32, 0, stream>>>(s, Wv, bv, nullptr, vb);
    gemm512_k<2><<<gg, 32, 0, stream>>>(s, Wg, bg, nullptr, gb);

    pair_bias_k<<<Bc * Nc * (Nc / 16), 32, 0, stream>>>(z, lg, lb, Wp, pb);

    attn_k<<<Bc * Hc * (Nc / 16), 32, 0, stream>>>(qb, kb, vb, pb, mask, ob);

    gemm512_k<3><<<gg, 32, 0, stream>>>(ob, Wo, bo, gb, out);
}